// SetAttentionLayer_34978213659074
// MI455X (gfx1250) — compile-verified
//
#include <hip/hip_runtime.h>
#include <math.h>

#define D_IN 16
#define WIDTH 128
#define LATENT 128
#define DOT 64
#define HEADS 4
#define N_SEG 1024
#define PAD 132   // LDS row stride (floats): (132%64)=4 -> conflict-free frag reads

typedef __attribute__((ext_vector_type(2))) float v2f;
typedef __attribute__((ext_vector_type(8))) float v8f;

__device__ __forceinline__ v8f wmma_f32(v2f a, v2f b, v8f c) {
  // D = A(16x4,f32) * B(4x16,f32) + C(16x16,f32)
  return __builtin_amdgcn_wmma_f32_16x16x4_f32(false, a, false, b, (short)0, c,
                                               false, false);
}

// ---------------------------------------------------------------- zero ws
__global__ void zero_kernel(unsigned* __restrict__ p, int count) {
  int i = blockIdx.x * 256 + threadIdx.x;
  if (i < count) p[i] = 0u;
}

// ------------------------------------------- contiguous segment boundaries
__global__ void bounds_kernel(const int* __restrict__ seg,
                              int* __restrict__ seg_start,
                              int* __restrict__ seg_end, int n) {
  int i = blockIdx.x * blockDim.x + threadIdx.x;
  if (i >= n) return;
  int s = seg[i];
  if (i == 0 || seg[i - 1] != s) seg_start[s] = i;
  if (i == n - 1 || seg[i + 1] != s) seg_end[s] = i + 1;
}

// ------------- repack weight [K x 128] into WMMA B-fragment order --------
// out[(kc*8 + ct)*32 + lane] = { W[4kc+2*half][ct*16+m], W[4kc+2*half+1][..] }
__global__ void pack_kernel(const float* __restrict__ W, v2f* __restrict__ out,
                            int kchunks) {
  int idx = blockIdx.x * 256 + threadIdx.x;
  if (idx >= kchunks * 8 * 32) return;
  int lane = idx & 31, rest = idx >> 5;
  int ct = rest & 7, kc = rest >> 3;
  int half = lane >> 4, m = lane & 15;
  int col = ct * 16 + m;
  int kr = 4 * kc + 2 * half;
  v2f v;
  v.x = W[kr * WIDTH + col];
  v.y = W[(kr + 1) * WIDTH + col];
  out[idx] = v;
}

// ---------------------------------------------- heavy: MLP + segment sums
// 2 waves/block, 2 row-tiles (32 rows) per wave -> 64 rows per block.
__global__ __launch_bounds__(64)
void mlp_seg_kernel(const float* __restrict__ in, const int* __restrict__ seg,
                    const v2f* __restrict__ Wp0, const float* __restrict__ b0,
                    const v2f* __restrict__ Wp1, const float* __restrict__ b1,
                    const v2f* __restrict__ Wpl, const float* __restrict__ bl,
                    float* __restrict__ seg_sum, int n) {
  __shared__ float lds_h[2][32 * PAD];
  const int lane = threadIdx.x & 31;
  const int wave = threadIdx.x >> 5;
  const int half = lane >> 4;   // 0: lanes 0-15, 1: lanes 16-31
  const int m = lane & 15;
  const int r0 = blockIdx.x * 64 + wave * 32;
  float* hb = lds_h[wave];

  v8f acc[2][8];
#pragma unroll
  for (int t = 0; t < 2; ++t)
#pragma unroll
    for (int ct = 0; ct < 8; ++ct)
#pragma unroll
      for (int r = 0; r < 8; ++r) acc[t][ct][r] = 0.f;

  // ---- layer 1: relu(X[16x16] @ W0 + b0), K=16 in 4 chunks
  v2f aIn[2][4];
#pragma unroll
  for (int t = 0; t < 2; ++t) {
    int row = r0 + t * 16 + m;
    if (row >= n) row = n - 1;                 // clamp (masked at accumulation)
    const float* rp = in + row * D_IN + 2 * half;
#pragma unroll
    for (int k = 0; k < 4; ++k) aIn[t][k] = *(const v2f*)(rp + 4 * k);
  }
#pragma unroll
  for (int k = 0; k < 4; ++k) {
#pragma unroll
    for (int ct = 0; ct < 8; ++ct) {
      v2f b = Wp0[(k * 8 + ct) * 32 + lane];
#pragma unroll
      for (int t = 0; t < 2; ++t) acc[t][ct] = wmma_f32(aIn[t][k], b, acc[t][ct]);
    }
  }
  // bias + relu -> LDS (C layout: reg r, row = t*16 + r + 8*half, col = ct*16+m)
#pragma unroll
  for (int ct = 0; ct < 8; ++ct) {
    float bv = b0[ct * 16 + m];
#pragma unroll
    for (int t = 0; t < 2; ++t)
#pragma unroll
      for (int r = 0; r < 8; ++r)
        hb[(t * 16 + r + 8 * half) * PAD + ct * 16 + m] =
            fmaxf(acc[t][ct][r] + bv, 0.f);
  }
  __syncthreads();

  // ---- layer 2: relu(H @ W1 + b1), K=128 in 32 chunks
#pragma unroll
  for (int t = 0; t < 2; ++t)
#pragma unroll
    for (int ct = 0; ct < 8; ++ct)
#pragma unroll
      for (int r = 0; r < 8; ++r) acc[t][ct][r] = 0.f;
  for (int k = 0; k < 32; ++k) {
    int kc = 4 * k + 2 * half;
    v2f a0 = *(const v2f*)(hb + m * PAD + kc);
    v2f a1 = *(const v2f*)(hb + (16 + m) * PAD + kc);
#pragma unroll
    for (int ct = 0; ct < 8; ++ct) {
      v2f b = Wp1[(k * 8 + ct) * 32 + lane];
      acc[0][ct] = wmma_f32(a0, b, acc[0][ct]);
      acc[1][ct] = wmma_f32(a1, b, acc[1][ct]);
    }
  }
  __syncthreads();
#pragma unroll
  for (int ct = 0; ct < 8; ++ct) {
    float bv = b1[ct * 16 + m];
#pragma unroll
    for (int t = 0; t < 2; ++t)
#pragma unroll
      for (int r = 0; r < 8; ++r)
        hb[(t * 16 + r + 8 * half) * PAD + ct * 16 + m] =
            fmaxf(acc[t][ct][r] + bv, 0.f);
  }
  __syncthreads();

  // ---- layer 3: enc = relu(H2 @ Wlat + blat)
#pragma unroll
  for (int t = 0; t < 2; ++t)
#pragma unroll
    for (int ct = 0; ct < 8; ++ct)
#pragma unroll
      for (int r = 0; r < 8; ++r) acc[t][ct][r] = 0.f;
  for (int k = 0; k < 32; ++k) {
    int kc = 4 * k + 2 * half;
    v2f a0 = *(const v2f*)(hb + m * PAD + kc);
    v2f a1 = *(const v2f*)(hb + (16 + m) * PAD + kc);
#pragma unroll
    for (int ct = 0; ct < 8; ++ct) {
      v2f b = Wpl[(k * 8 + ct) * 32 + lane];
      acc[0][ct] = wmma_f32(a0, b, acc[0][ct]);
      acc[1][ct] = wmma_f32(a1, b, acc[1][ct]);
    }
  }
#pragma unroll
  for (int ct = 0; ct < 8; ++ct) {
    float bv = bl[ct * 16 + m];
#pragma unroll
    for (int t = 0; t < 2; ++t)
#pragma unroll
      for (int r = 0; r < 8; ++r) acc[t][ct][r] = fmaxf(acc[t][ct][r] + bv, 0.f);
  }

  // ---- segment accumulation (per 16-row tile)
#pragma unroll
  for (int t = 0; t < 2; ++t) {
    int rt = r0 + t * 16;
    int segFirst = (rt < n) ? seg[rt] : -1;
    int segLast = (rt + 15 < n) ? seg[rt + 15] : -2;
    if (segFirst == segLast) {
      // fast path: whole tile in one segment -> column sums, 1 atomic/col
#pragma unroll
      for (int ct = 0; ct < 8; ++ct) {
        float cs = 0.f;
#pragma unroll
        for (int r = 0; r < 8; ++r) cs += acc[t][ct][r];
        cs += __shfl_xor(cs, 16);
        if (lane < 16) atomicAdd(&seg_sum[segFirst * WIDTH + ct * 16 + m], cs);
      }
    } else {
#pragma unroll
      for (int r = 0; r < 8; ++r) {
        int row = rt + r + 8 * half;
        if (row < n) {
          int sg = seg[row];
#pragma unroll
          for (int ct = 0; ct < 8; ++ct)
            atomicAdd(&seg_sum[sg * WIDTH + ct * 16 + m], acc[t][ct][r]);
        }
      }
    }
  }
}

// ----------------------------------------------------------- rho (tiny)
__global__ __launch_bounds__(128)
void rho_kernel(const float* __restrict__ seg_sum,
                const int* __restrict__ seg_start,
                const int* __restrict__ seg_end,
                const float* __restrict__ Wr, const float* __restrict__ br,
                float* __restrict__ agg) {
  int s = blockIdx.x, t = threadIdx.x;
  __shared__ float x[WIDTH];
  float cnt = fmaxf((float)(seg_end[s] - seg_start[s]), 1.f);
  x[t] = seg_sum[s * WIDTH + t] / cnt;
  __syncthreads();
  float a = br[t];
  for (int k = 0; k < WIDTH; ++k) a += x[k] * Wr[k * WIDTH + t];
  agg[s * WIDTH + t] = fmaxf(a, 0.f);
}

// -------------------------------- fold W_k with W_q (and 1/sqrt(DOT))
__global__ void weff_kernel(const float* __restrict__ Wk,
                            const float* __restrict__ Wq,
                            float* __restrict__ w_eff) {
  int idx = blockIdx.x * blockDim.x + threadIdx.x;
  if (idx >= (D_IN + LATENT) * HEADS) return;
  int c = idx >> 2, h = idx & 3;
  float sv = 0.f;
  for (int d = 0; d < DOT; ++d)
    sv += Wk[c * (DOT * HEADS) + h * DOT + d] * Wq[h * DOT + d];
  w_eff[idx] = sv * 0.125f;  // 1/sqrt(64)
}

// ------------------------------ per-segment constant part of the logits
__global__ void segbias_kernel(const float* __restrict__ agg,
                               const float* __restrict__ w_eff,
                               float* __restrict__ seg_bias) {
  int idx = blockIdx.x * blockDim.x + threadIdx.x;
  if (idx >= N_SEG * HEADS) return;
  int s = idx >> 2, h = idx & 3;
  float a = 0.f;
  for (int k = 0; k < LATENT; ++k)
    a += agg[s * LATENT + k] * w_eff[(D_IN + k) * HEADS + h];
  seg_bias[idx] = a;
}

// --------------------------------------------- logits pre[n,h]
__global__ __launch_bounds__(256)
void pre_kernel(const float* __restrict__ in, const int* __restrict__ seg,
                const float* __restrict__ w_eff,
                const float* __restrict__ seg_bias, float* __restrict__ pre,
                int n) {
  __shared__ float w[D_IN * HEADS];
  if (threadIdx.x < D_IN * HEADS) w[threadIdx.x] = w_eff[threadIdx.x];
  __syncthreads();
  int i = blockIdx.x * 256 + threadIdx.x;
  if (i >= n) return;
  const float4* rp = (const float4*)(in + i * D_IN);
  float4 q0 = rp[0], q1 = rp[1], q2 = rp[2], q3 = rp[3];
  float x[16] = {q0.x, q0.y, q0.z, q0.w, q1.x, q1.y, q1.z, q1.w,
                 q2.x, q2.y, q2.z, q2.w, q3.x, q3.y, q3.z, q3.w};
  int s = seg[i];
#pragma unroll
  for (int h = 0; h < HEADS; ++h) {
    float p = seg_bias[s * HEADS + h];
#pragma unroll
    for (int c = 0; c < D_IN; ++c) p += x[c] * w[c * HEADS + h];
    pre[i * HEADS + h] = p;
  }
}

// ------------------------------- per-segment, per-head softmax (3 passes)
__global__ __launch_bounds__(256)
void softmax_kernel(const float* __restrict__ pre,
                    const int* __restrict__ seg_start,
                    const int* __restrict__ seg_end, float* __restrict__ out) {
  int s = blockIdx.x;
  int st = seg_start[s], en = seg_end[s];
  int t = threadIdx.x, lane = t & 31, wv = t >> 5;
  __shared__ float red[8 * HEADS];
  __shared__ float bmax[HEADS];
  __shared__ float bsum[HEADS];

  float mx[HEADS] = {-INFINITY, -INFINITY, -INFINITY, -INFINITY};
  for (int i = st + t; i < en; i += 256)
#pragma unroll
    for (int h = 0; h < HEADS; ++h) mx[h] = fmaxf(mx[h], pre[i * HEADS + h]);
#pragma unroll
  for (int off = 16; off; off >>= 1)
#pragma unroll
    for (int h = 0; h < HEADS; ++h)
      mx[h] = fmaxf(mx[h], __shfl_xor(mx[h], off));
  if (lane == 0)
#pragma unroll
    for (int h = 0; h < HEADS; ++h) red[wv * HEADS + h] = mx[h];
  __syncthreads();
  if (t < HEADS) {
    float v = red[t];
    for (int w2 = 1; w2 < 8; ++w2) v = fmaxf(v, red[w2 * HEADS + t]);
    bmax[t] = v;
  }
  __syncthreads();

  float sm[HEADS] = {0.f, 0.f, 0.f, 0.f};
  for (int i = st + t; i < en; i += 256)
#pragma unroll
    for (int h = 0; h < HEADS; ++h) sm[h] += expf(pre[i * HEADS + h] - bmax[h]);
#pragma unroll
  for (int off = 16; off; off >>= 1)
#pragma unroll
    for (int h = 0; h < HEADS; ++h) sm[h] += __shfl_xor(sm[h], off);
  if (lane == 0)
#pragma unroll
    for (int h = 0; h < HEADS; ++h) red[wv * HEADS + h] = sm[h];
  __syncthreads();
  if (t < HEADS) {
    float v = 0.f;
    for (int w2 = 0; w2 < 8; ++w2) v += red[w2 * HEADS + t];
    bsum[t] = v;
  }
  __syncthreads();

  float inv[HEADS];
#pragma unroll
  for (int h = 0; h < HEADS; ++h) inv[h] = 1.f / bsum[h];
  for (int i = st + t; i < en; i += 256)
#pragma unroll
    for (int h = 0; h < HEADS; ++h)
      out[i * HEADS + h] = expf(pre[i * HEADS + h] - bmax[h]) * inv[h];
}

extern "C" void kernel_launch(void* const* d_in, const int* in_sizes, int n_in,
                              void* d_out, int out_size, void* d_ws,
                              size_t ws_size, hipStream_t stream) {
  const float* in = (const float*)d_in[0];
  const int* seg = (const int*)d_in[1];
  // d_in[2] = lengths: unused by the reference computation
  const float* W0 = (const float*)d_in[3];
  const float* b0 = (const float*)d_in[4];
  const float* W1 = (const float*)d_in[5];
  const float* b1 = (const float*)d_in[6];
  const float* Wl = (const float*)d_in[7];
  const float* bl = (const float*)d_in[8];
  const float* Wr = (const float*)d_in[9];
  const float* br = (const float*)d_in[10];
  const float* Wk = (const float*)d_in[11];
  const float* Wq = (const float*)d_in[12];
  const int n = in_sizes[0] / D_IN;

  char* ws = (char*)d_ws;
  float* seg_sum = (float*)ws;  ws += (size_t)N_SEG * WIDTH * 4;
  int* seg_start = (int*)ws;    ws += (size_t)N_SEG * 4;
  int* seg_end = (int*)ws;      ws += (size_t)N_SEG * 4;
  float* agg = (float*)ws;      ws += (size_t)N_SEG * LATENT * 4;
  float* w_eff = (float*)ws;    ws += (size_t)(D_IN + LATENT) * HEADS * 4;
  float* seg_bias = (float*)ws; ws += (size_t)N_SEG * HEADS * 4;
  v2f* Wp0 = (v2f*)ws;          ws += (size_t)4 * 8 * 32 * 8;    // 8 KB
  v2f* Wp1 = (v2f*)ws;          ws += (size_t)32 * 8 * 32 * 8;   // 64 KB
  v2f* Wpl = (v2f*)ws;          ws += (size_t)32 * 8 * 32 * 8;   // 64 KB
  float* pre = (float*)ws;      // N * HEADS floats

  // 1) zero the accumulator region (seg_sum + seg_start + seg_end contiguous)
  int zwords = N_SEG * WIDTH + 2 * N_SEG;
  zero_kernel<<<(zwords + 255) / 256, 256, 0, stream>>>((unsigned*)seg_sum,
                                                        zwords);
  // 2) segment boundaries
  bounds_kernel<<<(n + 255) / 256, 256, 0, stream>>>(seg, seg_start, seg_end, n);
  // 3) repack weights into WMMA B-fragment order (coalesced b64 fetches)
  pack_kernel<<<(4 * 8 * 32 + 255) / 256, 256, 0, stream>>>(W0, Wp0, 4);
  pack_kernel<<<(32 * 8 * 32 + 255) / 256, 256, 0, stream>>>(W1, Wp1, 32);
  pack_kernel<<<(32 * 8 * 32 + 255) / 256, 256, 0, stream>>>(Wl, Wpl, 32);
  // 4) heavy MLP + segment-sum (WMMA path), 64 rows per block
  mlp_seg_kernel<<<(n + 63) / 64, 64, 0, stream>>>(in, seg, Wp0, b0, Wp1, b1,
                                                   Wpl, bl, seg_sum, n);
  // 5) rho on segment means
  rho_kernel<<<N_SEG, 128, 0, stream>>>(seg_sum, seg_start, seg_end, Wr, br, agg);
  // 6) fold W_k/W_q, per-segment logit bias
  weff_kernel<<<((D_IN + LATENT) * HEADS + 255) / 256, 256, 0, stream>>>(Wk, Wq,
                                                                         w_eff);
  segbias_kernel<<<(N_SEG * HEADS + 255) / 256, 256, 0, stream>>>(agg, w_eff,
                                                                  seg_bias);
  // 7) logits
  pre_kernel<<<(n + 255) / 256, 256, 0, stream>>>(in, seg, w_eff, seg_bias, pre,
                                                  n);
  // 8) per-segment softmax -> attn [N, HEADS]
  softmax_kernel<<<N_SEG, 256, 0, stream>>>(pre, seg_start, seg_end,
                                            (float*)d_out);
}